// CoordinatePredictor_68908455297211
// MI455X (gfx1250) — compile-verified
//
#include <hip/hip_runtime.h>

typedef __attribute__((ext_vector_type(2))) float v2f;
typedef __attribute__((ext_vector_type(8))) float v8f;

#define DF 128     // feature dim (D_IN == D_HID == 128)
#define LDSP 132   // padded LDS row stride: 16B-aligned rows; %64=4 -> conflict-free frag reads

// ---------------------------------------------------------------------------
// zero fill for int buffers (4 per thread)
// ---------------------------------------------------------------------------
__global__ __launch_bounds__(256) void zero_int_kernel(int* __restrict__ p, int n) {
  int i = (blockIdx.x * blockDim.x + threadIdx.x) * 4;
  if (i + 3 < n) {
    *reinterpret_cast<int4*>(p + i) = make_int4(0, 0, 0, 0);
  } else {
    for (int j = i; j < n; ++j) p[j] = 0;
  }
}

// ---------------------------------------------------------------------------
// degree histogram over (edges + self loops)
// ---------------------------------------------------------------------------
__global__ __launch_bounds__(256) void count_kernel(const int* __restrict__ col,
                                                    int* __restrict__ deg,
                                                    int E, int N) {
  int e = blockIdx.x * blockDim.x + threadIdx.x;
  if (e < E) {
    atomicAdd(&deg[col[e]], 1);
  } else if (e < E + N) {
    atomicAdd(&deg[e - E], 1);  // self loop
  }
}

// ---------------------------------------------------------------------------
// single-block exclusive scan of deg -> colStart[N+1]; fused deg_inv_sqrt
// ---------------------------------------------------------------------------
__global__ __launch_bounds__(1024) void scan_kernel(const int* __restrict__ deg,
                                                    int* __restrict__ colStart,
                                                    float* __restrict__ dis,
                                                    int N) {
  __shared__ int sdata[1024];
  __shared__ int sbase;
  const int tid = threadIdx.x;
  if (tid == 0) sbase = 0;
  __syncthreads();
  for (int base = 0; base < N; base += 1024) {
    const int i = base + tid;
    const int v = (i < N) ? deg[i] : 0;
    if (i < N) dis[i] = (v > 0) ? rsqrtf((float)v) : 0.f;
    sdata[tid] = v;
    __syncthreads();
    // Hillis-Steele inclusive scan
    for (int off = 1; off < 1024; off <<= 1) {
      const int t = (tid >= off) ? sdata[tid - off] : 0;
      __syncthreads();
      sdata[tid] += t;
      __syncthreads();
    }
    if (i < N) colStart[i] = sbase + sdata[tid] - v;  // exclusive
    __syncthreads();
    if (tid == 1023) sbase += sdata[1023];
    __syncthreads();
  }
  if (tid == 0) colStart[N] = sbase;  // == E + N
}

// ---------------------------------------------------------------------------
// bucket-fill: adjEid[colStart[c] + pos] = edge id (self loops have id E+node)
// ---------------------------------------------------------------------------
__global__ __launch_bounds__(256) void fill_kernel(const int* __restrict__ col,
                                                   const int* __restrict__ colStart,
                                                   int* __restrict__ cursor,
                                                   int* __restrict__ adjEid,
                                                   int E, int N) {
  int e = blockIdx.x * blockDim.x + threadIdx.x;
  if (e >= E + N) return;
  const int c = (e < E) ? col[e] : (e - E);
  const int pos = atomicAdd(&cursor[c], 1);
  adjEid[colStart[c] + pos] = e;
}

// ---------------------------------------------------------------------------
// per-node insertion sort of the edge-id segment -> deterministic sum order
// ---------------------------------------------------------------------------
__global__ __launch_bounds__(256) void sort_kernel(const int* __restrict__ colStart,
                                                   int* __restrict__ adjEid,
                                                   int N) {
  int c = blockIdx.x * blockDim.x + threadIdx.x;
  if (c >= N) return;
  const int s = colStart[c];
  const int e = colStart[c + 1];
  for (int i = s + 1; i < e; ++i) {
    const int key = adjEid[i];
    int j = i - 1;
    while (j >= s && adjEid[j] > key) {
      adjEid[j + 1] = adjEid[j];
      --j;
    }
    adjEid[j + 1] = key;
  }
}

// ---------------------------------------------------------------------------
// WMMA fp32 GEMM: Out[nrows x 128] = act(A + bias) @ W
//   FUSE=false: act = identity (layer-1 input x)
//   FUSE=true : act = relu(A + bias) (previous conv's bias, fused)
// Block = 256 threads = 8 waves, 128-row strip; A tile + W in padded LDS.
// ---------------------------------------------------------------------------
template <bool FUSE>
__global__ __launch_bounds__(256) void gemm_kernel(const float* __restrict__ A,
                                                   const float* __restrict__ W,
                                                   const float* __restrict__ bias,
                                                   float* __restrict__ Out,
                                                   int nrows) {
  __shared__ float sA[DF * LDSP];
  __shared__ float sW[DF * LDSP];
  const int tid = threadIdx.x;
  const int mbase = blockIdx.x * DF;

#pragma unroll
  for (int it = 0; it < 16; ++it) {
    const int lin = it * 1024 + tid * 4;
    const int r = lin >> 7;
    const int c = lin & 127;
    *reinterpret_cast<float4*>(sW + r * LDSP + c) =
        *reinterpret_cast<const float4*>(W + r * DF + c);
    const int gr = mbase + r;
    float4 a4 = make_float4(0.f, 0.f, 0.f, 0.f);
    if (gr < nrows) {
      a4 = *reinterpret_cast<const float4*>(A + (size_t)gr * DF + c);
      if (FUSE) {
        const float4 b4 = *reinterpret_cast<const float4*>(bias + c);
        a4.x = fmaxf(a4.x + b4.x, 0.f);
        a4.y = fmaxf(a4.y + b4.y, 0.f);
        a4.z = fmaxf(a4.z + b4.z, 0.f);
        a4.w = fmaxf(a4.w + b4.w, 0.f);
      }
    }
    *reinterpret_cast<float4*>(sA + r * LDSP + c) = a4;
  }
  __syncthreads();

  const int wave = tid >> 5;
  const int lane = tid & 31;
  const int l16  = lane & 15;
  const int half = lane >> 4;
  const int arow = (wave * 16 + l16) * LDSP;

  for (int n0 = 0; n0 < DF; n0 += 16) {
    v8f acc = {};
#pragma unroll
    for (int k0 = 0; k0 < DF; k0 += 4) {
      const int kk = k0 + 2 * half;
      v2f a, b;
      a[0] = sA[arow + kk];
      a[1] = sA[arow + kk + 1];
      b[0] = sW[kk * LDSP + n0 + l16];
      b[1] = sW[(kk + 1) * LDSP + n0 + l16];
      acc = __builtin_amdgcn_wmma_f32_16x16x4_f32(false, a, false, b,
                                                  (short)0, acc, false, false);
    }
#pragma unroll
    for (int v = 0; v < 8; ++v) {
      const int gr = mbase + wave * 16 + v + 8 * half;
      if (gr < nrows) Out[(size_t)gr * DF + n0 + l16] = acc[v];
    }
  }
}

// ---------------------------------------------------------------------------
// CSC gather-aggregate: one wave per destination node, register accumulation,
// no atomics, single coalesced 512B store per node.
//   agg[c] = sum_{e in in(c)} h[row(e)] * dis[row(e)] * dis[c]
// ---------------------------------------------------------------------------
__global__ __launch_bounds__(256) void gather_kernel(const float* __restrict__ h,
                                                     const int* __restrict__ row,
                                                     const int* __restrict__ adjEid,
                                                     const int* __restrict__ colStart,
                                                     const float* __restrict__ dis,
                                                     float* __restrict__ agg,
                                                     int E, int N) {
  const int c = (int)(((long long)blockIdx.x * blockDim.x + threadIdx.x) >> 5);
  const int lane = threadIdx.x & 31;
  if (c >= N) return;
  const float dc = dis[c];
  const int s = colStart[c];
  const int e = colStart[c + 1];
  float4 acc = make_float4(0.f, 0.f, 0.f, 0.f);
  for (int i = s; i < e; ++i) {
    const int eid = adjEid[i];
    const int r = (eid < E) ? row[eid] : (eid - E);
    const float w = dis[r] * dc;
    const float4 v = *reinterpret_cast<const float4*>(h + (size_t)r * DF + lane * 4);
    acc.x += v.x * w;
    acc.y += v.y * w;
    acc.z += v.z * w;
    acc.w += v.w * w;
  }
  *reinterpret_cast<float4*>(agg + (size_t)c * DF + lane * 4) = acc;
}

// ---------------------------------------------------------------------------
// head: out[N x 3] = relu(agg + b2) @ Wh + bh ; one wave per node
// ---------------------------------------------------------------------------
__global__ __launch_bounds__(256) void head_kernel(const float* __restrict__ agg,
                                                   const float* __restrict__ b2,
                                                   const float* __restrict__ Wh,
                                                   const float* __restrict__ bh,
                                                   float* __restrict__ out,
                                                   int N) {
  const int gw = (int)(((long long)blockIdx.x * blockDim.x + threadIdx.x) >> 5);
  const int lane = threadIdx.x & 31;
  if (gw >= N) return;
  const float* __restrict__ rp = agg + (size_t)gw * DF;
  float s0 = 0.f, s1 = 0.f, s2 = 0.f;
#pragma unroll
  for (int j = 0; j < 4; ++j) {
    const int k = lane + j * 32;
    float a = rp[k] + b2[k];
    a = a > 0.f ? a : 0.f;
    s0 += a * Wh[k * 3 + 0];
    s1 += a * Wh[k * 3 + 1];
    s2 += a * Wh[k * 3 + 2];
  }
#pragma unroll
  for (int off = 16; off > 0; off >>= 1) {
    s0 += __shfl_down(s0, off, 32);
    s1 += __shfl_down(s1, off, 32);
    s2 += __shfl_down(s2, off, 32);
  }
  if (lane == 0) {
    out[(size_t)gw * 3 + 0] = s0 + bh[0];
    out[(size_t)gw * 3 + 1] = s1 + bh[1];
    out[(size_t)gw * 3 + 2] = s2 + bh[2];
  }
}

// ---------------------------------------------------------------------------
// launcher — inputs: x, edge_index, W1, b1, W2, b2, Wh, bh
// ---------------------------------------------------------------------------
extern "C" void kernel_launch(void* const* d_in, const int* in_sizes, int n_in,
                              void* d_out, int out_size, void* d_ws, size_t ws_size,
                              hipStream_t stream) {
  const float* x  = (const float*)d_in[0];
  const int*   ei = (const int*)d_in[1];
  const float* W1 = (const float*)d_in[2];
  const float* b1 = (const float*)d_in[3];
  const float* W2 = (const float*)d_in[4];
  const float* b2 = (const float*)d_in[5];
  const float* Wh = (const float*)d_in[6];
  const float* bh = (const float*)d_in[7];
  float* out = (float*)d_out;

  const int N = in_sizes[0] / DF;
  const int E = in_sizes[1] / 2;
  const int* row = ei;       // edge_index[0]
  const int* col = ei + E;   // edge_index[1]
  const int M = E + N;       // edges incl. self loops

  // workspace layout (all 256B aligned)
  char* ws = (char*)d_ws;
  size_t off = 0;
  auto alloc = [&](size_t bytes) {
    void* p = ws + off;
    off += (bytes + 255) & ~(size_t)255;
    return p;
  };
  float* dis      = (float*)alloc((size_t)N * 4);
  int*   degInt   = (int*)alloc((size_t)N * 4);
  int*   colStart = (int*)alloc(((size_t)N + 1) * 4);
  int*   cursor   = (int*)alloc((size_t)N * 4);
  int*   adjEid   = (int*)alloc((size_t)M * 4);
  float* bufA     = (float*)alloc((size_t)N * DF * 4);
  float* bufB     = (float*)alloc((size_t)N * DF * 4);

  const int gemmBlocks = (N + DF - 1) / DF;
  const int waveNBlocks = (int)(((long long)N * 32 + 255) / 256);
  const int edgeBlocks = (M + 255) / 256;
  const int zeroNBlocks = (N / 4 + 255) / 256 + 1;

  // ---- build deterministic CSC (deg -> scan -> fill -> sort) + deg_inv_sqrt
  zero_int_kernel<<<zeroNBlocks, 256, 0, stream>>>(degInt, N);
  zero_int_kernel<<<zeroNBlocks, 256, 0, stream>>>(cursor, N);
  count_kernel<<<edgeBlocks, 256, 0, stream>>>(col, degInt, E, N);
  scan_kernel<<<1, 1024, 0, stream>>>(degInt, colStart, dis, N);
  fill_kernel<<<edgeBlocks, 256, 0, stream>>>(col, colStart, cursor, adjEid, E, N);
  sort_kernel<<<(N + 255) / 256, 256, 0, stream>>>(colStart, adjEid, N);

  // ---- conv1: h1 = x @ W1 ; agg1 = A_norm @ h1
  gemm_kernel<false><<<gemmBlocks, 256, 0, stream>>>(x, W1, b1, bufA, N);
  gather_kernel<<<waveNBlocks, 256, 0, stream>>>(bufA, row, adjEid, colStart, dis,
                                                 bufB, E, N);

  // ---- conv2: h2 = relu(agg1 + b1) @ W2 ; agg2 = A_norm @ h2
  gemm_kernel<true><<<gemmBlocks, 256, 0, stream>>>(bufB, W2, b1, bufA, N);
  gather_kernel<<<waveNBlocks, 256, 0, stream>>>(bufA, row, adjEid, colStart, dis,
                                                 bufB, E, N);

  // ---- head: out = relu(agg2 + b2) @ Wh + bh
  head_kernel<<<waveNBlocks, 256, 0, stream>>>(bufB, b2, Wh, bh, out, N);
}